// LogitLayer_83562883711883
// MI455X (gfx1250) — compile-verified
//
#include <hip/hip_runtime.h>

// out[e] = exp(-r * (nc[i] - nc[j] + v[e])), e in [0, NNZ)
//
// Roofline (MI455X): 24 B/edge streamed from HBM (4B value + 16B int64 pair +
// 4B out) -> ~480 MB total -> ~20.6 us floor at 23.3 TB/s. node_constants is
// 400 KB: fully L2-resident (192 MB), so gathers are L2 hits after warmup.
// Compute (2 VALU + 1 exp per edge) is far below the VALU ceiling -> purely
// memory-bound; no matrix semantics, so no WMMA by design.
//
// Strategy: 4 edges/thread; all streaming traffic as 128-bit non-temporal
// vector ops (preserve L2 for the gather table); gfx1250 global_prefetch_b8
// to pre-warm the table into L2; 32-bit offsets so loads use SGPR64-base +
// VGPR32-offset (GVS) addressing.

typedef float     v4f   __attribute__((ext_vector_type(4)));
typedef long long v2i64 __attribute__((ext_vector_type(2)));

__global__ __launch_bounds__(256) void logit_edge_kernel(
    const float* __restrict__ values,
    const float* __restrict__ node_constants,
    const float* __restrict__ rationality,
    const long long* __restrict__ indices,   // [nnz][2] int64
    float* __restrict__ out,
    int nnz,
    int n_nodes)
{
    const int gid = (int)(blockIdx.x * blockDim.x + threadIdx.x);   // < 5M, fits int

    // Warm the gather table into L2: one 128 B line (32 floats) per early
    // thread; 3125 prefetches cover all 100k floats. -> global_prefetch_b8
    const int warm_lines = (n_nodes + 31) >> 5;
    if (gid < warm_lines) {
        __builtin_prefetch(node_constants + (gid << 5), 0, 3);
    }

    const float negR = -rationality[0];   // uniform scalar load

    const int e0 = gid << 2;              // max 20M, fits int
    if (e0 >= nnz) return;

    if (e0 + 3 < nnz) {
        // ---- fast path: 4 edges, all streaming as b128 non-temporal ----
        const v4f v = __builtin_nontemporal_load((const v4f*)(values + e0));

        const v2i64 p0 = __builtin_nontemporal_load((const v2i64*)(indices + 2 * (e0 + 0)));
        const v2i64 p1 = __builtin_nontemporal_load((const v2i64*)(indices + 2 * (e0 + 1)));
        const v2i64 p2 = __builtin_nontemporal_load((const v2i64*)(indices + 2 * (e0 + 2)));
        const v2i64 p3 = __builtin_nontemporal_load((const v2i64*)(indices + 2 * (e0 + 3)));

        // node ids < 100000 and non-negative -> low 32 bits suffice
        const float c0 = node_constants[(int)p0.x] - node_constants[(int)p0.y];
        const float c1 = node_constants[(int)p1.x] - node_constants[(int)p1.y];
        const float c2 = node_constants[(int)p2.x] - node_constants[(int)p2.y];
        const float c3 = node_constants[(int)p3.x] - node_constants[(int)p3.y];

        v4f o;
        o.x = expf(negR * (c0 + v.x));
        o.y = expf(negR * (c1 + v.y));
        o.z = expf(negR * (c2 + v.z));
        o.w = expf(negR * (c3 + v.w));

        __builtin_nontemporal_store(o, (v4f*)(out + e0));
    } else {
        // ---- tail: scalar (at most 3 edges; unreachable when 4 | nnz) ----
        for (int e = e0; e < nnz; ++e) {
            const int i = (int)indices[2 * e + 0];
            const int j = (int)indices[2 * e + 1];
            const float c = node_constants[i] - node_constants[j];
            out[e] = expf(negR * (c + values[e]));
        }
    }
}

extern "C" void kernel_launch(void* const* d_in, const int* in_sizes, int n_in,
                              void* d_out, int out_size, void* d_ws, size_t ws_size,
                              hipStream_t stream) {
    const float*     values         = (const float*)d_in[0];
    const float*     node_constants = (const float*)d_in[1];
    const float*     rationality    = (const float*)d_in[2];
    const long long* indices        = (const long long*)d_in[3];
    float*           out            = (float*)d_out;

    const int nnz     = in_sizes[0];      // 20,000,000
    const int n_nodes = in_sizes[1];      // 100,000

    const int block  = 256;                                   // 8 wave32s
    const int groups = (nnz + 3) / 4;                         // 5,000,000
    const int grid   = (groups + block - 1) / block;          // 19,532

    logit_edge_kernel<<<grid, block, 0, stream>>>(
        values, node_constants, rationality, indices, out, nnz, n_nodes);
}